// ConvLayerAcceleratedV1_84748294684828
// MI455X (gfx1250) — compile-verified
//
#include <hip/hip_runtime.h>
#include <hip/hip_bf16.h>
#include <math.h>

// ---------------------------------------------------------------------------
// MI455X / gfx1250: wave32, WMMA f32 16x16x4 matrix cores.
//  - every GEMM on v_wmma_f32_16x16x4_f32 (fp32 reference -> fp32 matrix ops)
//  - fused tensor-product + segment_sum via f32 atomics into a 19.2MB
//    accumulator that stays resident in the 192MB L2
//  - weights pre-transposed to fragment-major so each B fragment is one
//    global_load_b64; A fragments loaded as v2f and hoisted across all
//    column tiles (reused up to 14x in the edge MLP)
// ---------------------------------------------------------------------------

typedef __attribute__((ext_vector_type(2))) float v2f;
typedef __attribute__((ext_vector_type(8))) float v8f;

#define N_NODES 10000
#define N_EDGES 320000

#define INV_SQ3   0.5773502691896258f   // 1/sqrt(3)
#define INV_SQ2   0.7071067811865476f   // 1/sqrt(2)
#define INV_SQ32  0.17677669529663687f  // 1/sqrt(32)
#define INV_SQ64  0.125f                // 1/sqrt(64)
#define INV_SQ96  0.10206207261596575f  // 1/sqrt(96)
#define INV_SQ160 0.07905694150420949f  // 1/sqrt(160)
#define LN2F      0.6931471805599453f

__device__ __forceinline__ float sspf(float v) {
  float sp = (v > 20.0f) ? v : log1pf(expf(v));
  return sp - LN2F;
}
__device__ __forceinline__ float siluf(float v) {
  return v / (1.0f + expf(-v));
}
__device__ __forceinline__ v2f ld2(const float* p) {
  return *(const v2f*)p;  // callers guarantee 8B alignment (even float index)
}

__device__ __forceinline__ v8f wmma4(v2f a, v2f b, v8f c) {
  return __builtin_amdgcn_wmma_f32_16x16x4_f32(false, a, false, b,
                                               (short)0, c, false, false);
}

// Per ISA 7.12.2 (wave32), V_WMMA_F32_16X16X4_F32 fragments:
//   A: lane L -> row m = L&15, k-pair {kb+2*(L>>4), +1}
//   B: lane L -> col n = L&15, same k-pair
//   C/D: VGPR r -> row r + 8*(L>>4), col L&15

// Load KS k-step A fragments (16 x 4*KS panel) into registers once.
template <int KS, typename LA>
__device__ __forceinline__ void load_a(v2f (&a)[KS], LA la) {
  const int lane = threadIdx.x & 31;
  const int m = lane & 15;
  const int kk = (lane >> 4) << 1;
#pragma unroll
  for (int q = 0; q < KS; ++q) a[q] = la(m, q * 4 + kk);
}

// C += A_panel * B_tile with pre-loaded A fragments; lb(n, k) -> v2f {B[k][n],B[k+1][n]}.
template <int KS, typename LB>
__device__ __forceinline__ v8f wmma_tile(const v2f (&a)[KS], LB lb, v8f c) {
  const int lane = threadIdx.x & 31;
  const int n = lane & 15;
  const int kk = (lane >> 4) << 1;
#pragma unroll
  for (int q = 0; q < KS; ++q) c = wmma4(a[q], lb(n, q * 4 + kk), c);
  return c;
}

// st(r, row, col, value) for each of the 8 accumulator registers.
template <typename ST>
__device__ __forceinline__ void tile_store(v8f c, ST st) {
  const int lane = threadIdx.x & 31;
  const int col = lane & 15;
  const int rbase = (lane >> 4) << 3;
#pragma unroll
  for (int r = 0; r < 8; ++r) st(r, rbase + r, col, c[r]);
}

// ---------------------------------------------------------------------------
// Weight transpose: W(K x Nc) row-major -> Wt(Nc x K) so a B fragment
// {W[k][n], W[k+1][n]} is one contiguous 8-byte load.
// ---------------------------------------------------------------------------
__global__ void transpose_kernel(const float* __restrict__ W,
                                 float* __restrict__ Wt, int K, int Nc) {
  int idx = blockIdx.x * blockDim.x + threadIdx.x;
  if (idx < K * Nc) {
    int k = idx / Nc, n = idx % Nc;
    Wt[n * K + k] = W[idx];
  }
}

// ---------------------------------------------------------------------------
// Kernel 1: per-node precompute (one wave per 16-node tile).
//   p0  = x0 @ W0p/8 + b0p                  p1 = einsum(x1, W1p)/sqrt(32)
//   g   = silu(concat(x0,|x1|) @ G1 + g1b) @ G2 + g2b
//   x0n = g[:, :64] @ W0n/8 + b0n           x1n = einsum(x1*g[:,64:], W1n)/sqrt(32)
// All weight pointers here are the TRANSPOSED copies (Nc x K).
// ---------------------------------------------------------------------------
__global__ __launch_bounds__(32) void node_kernel(
    const float* __restrict__ x,
    const float* __restrict__ W0pT, const float* __restrict__ b0p,
    const float* __restrict__ W1pT,
    const float* __restrict__ W0nT, const float* __restrict__ b0n,
    const float* __restrict__ W1nT,
    const float* __restrict__ G1T, const float* __restrict__ g1b,
    const float* __restrict__ G2T, const float* __restrict__ g2b,
    float* __restrict__ p0, float* __restrict__ p1,
    float* __restrict__ x0n, float* __restrict__ x1n) {
  __shared__ __align__(16) float f0S[16 * 96];
  __shared__ __align__(16) float hS[16 * 96];
  __shared__ __align__(16) float gS[16 * 96];

  const int lane = threadIdx.x;
  const int n0 = blockIdx.x * 16;
  const float* xt = x + (size_t)n0 * 160;

  // p0 = x0 @ W0p/8 + b0p  (K=64, 4 column tiles share the A panel)
  {
    v2f ax[16];
    load_a(ax, [&](int m, int k) { return ld2(xt + m * 160 + k); });
    for (int j = 0; j < 4; ++j) {
      v8f c = {};
      c = wmma_tile(ax, [&](int n, int k) {
        return ld2(W0pT + (size_t)(j * 16 + n) * 64 + k);
      }, c);
      tile_store(c, [&](int, int m, int n, float v) {
        int col = j * 16 + n;
        p0[(size_t)(n0 + m) * 64 + col] = v * INV_SQ64 + b0p[col];
      });
    }
  }
  // p1[n,v,i] = sum_u x1[n,u,i] W1p[u,v] / sqrt(32)  (strided A, K=32)
  for (int i = 0; i < 3; ++i) {
    v2f ax[8];
    load_a(ax, [&](int m, int k) {
      v2f a;
      a.x = xt[m * 160 + 64 + 3 * k + i];
      a.y = xt[m * 160 + 64 + 3 * (k + 1) + i];
      return a;
    });
    for (int j = 0; j < 2; ++j) {
      v8f c = {};
      c = wmma_tile(ax, [&](int n, int k) {
        return ld2(W1pT + (size_t)(j * 16 + n) * 32 + k);
      }, c);
      tile_store(c, [&](int, int m, int n, float v) {
        p1[(size_t)(n0 + m) * 96 + (j * 16 + n) * 3 + i] = v * INV_SQ32;
      });
    }
  }

  // f0 = concat(x0, ||x1||)
  for (int idx = lane; idx < 16 * 96; idx += 32) {
    int m = idx / 96, c = idx % 96;
    float v;
    if (c < 64) {
      v = xt[m * 160 + c];
    } else {
      int u = c - 64;
      float a0 = xt[m * 160 + 64 + u * 3 + 0];
      float a1 = xt[m * 160 + 64 + u * 3 + 1];
      float a2 = xt[m * 160 + 64 + u * 3 + 2];
      v = sqrtf(a0 * a0 + a1 * a1 + a2 * a2 + 1e-12f);
    }
    f0S[idx] = v;
  }
  __syncthreads();

  // h = silu(f0 @ G1 + g1b)   (G1 pre-scaled in setup; K=96)
  {
    v2f af[24];
    load_a(af, [&](int m, int k) { return ld2(f0S + m * 96 + k); });
    for (int j = 0; j < 6; ++j) {
      v8f c = {};
      c = wmma_tile(af, [&](int n, int k) {
        return ld2(G1T + (size_t)(j * 16 + n) * 96 + k);
      }, c);
      tile_store(c, [&](int, int m, int n, float v) {
        int col = j * 16 + n;
        hS[m * 96 + col] = siluf(v + g1b[col]);
      });
    }
  }
  __syncthreads();
  // g = h @ G2 + g2b
  {
    v2f ah[24];
    load_a(ah, [&](int m, int k) { return ld2(hS + m * 96 + k); });
    for (int j = 0; j < 6; ++j) {
      v8f c = {};
      c = wmma_tile(ah, [&](int n, int k) {
        return ld2(G2T + (size_t)(j * 16 + n) * 96 + k);
      }, c);
      tile_store(c, [&](int, int m, int n, float v) {
        gS[m * 96 + j * 16 + n] = v + g2b[j * 16 + n];
      });
    }
  }
  __syncthreads();

  // x0n = g[:, :64] @ W0n/8 + b0n  (K=64)
  {
    v2f ag[16];
    load_a(ag, [&](int m, int k) { return ld2(gS + m * 96 + k); });
    for (int j = 0; j < 4; ++j) {
      v8f c = {};
      c = wmma_tile(ag, [&](int n, int k) {
        return ld2(W0nT + (size_t)(j * 16 + n) * 64 + k);
      }, c);
      tile_store(c, [&](int, int m, int n, float v) {
        int col = j * 16 + n;
        x0n[(size_t)(n0 + m) * 64 + col] = v * INV_SQ64 + b0n[col];
      });
    }
  }
  // x1n[n,v,i] = sum_u (x1[n,u,i]*g[n,64+u]) W1n[u,v] / sqrt(32)
  for (int i = 0; i < 3; ++i) {
    v2f ax[8];
    load_a(ax, [&](int m, int k) {
      v2f a;
      a.x = xt[m * 160 + 64 + 3 * k + i] * gS[m * 96 + 64 + k];
      a.y = xt[m * 160 + 64 + 3 * (k + 1) + i] * gS[m * 96 + 64 + k + 1];
      return a;
    });
    for (int j = 0; j < 2; ++j) {
      v8f c = {};
      c = wmma_tile(ax, [&](int n, int k) {
        return ld2(W1nT + (size_t)(j * 16 + n) * 32 + k);
      }, c);
      tile_store(c, [&](int, int m, int n, float v) {
        x1n[(size_t)(n0 + m) * 96 + (j * 16 + n) * 3 + i] = v * INV_SQ32;
      });
    }
  }
}

// ---------------------------------------------------------------------------
// Kernel 2: fused edge weight-MLPs + tensor product + segment_sum scatter.
// One wave per 16-edge tile. A panels (hN/hE) are loaded into registers ONCE
// and reused across all 14 column tiles of the 224-wide MLPs; atomics land
// in the L2-resident N x 480 accumulator.
// ---------------------------------------------------------------------------
__global__ __launch_bounds__(32) void edge_kernel(
    const float* __restrict__ edge_sh, const float* __restrict__ edge_attr,
    const float* __restrict__ F1T, const float* __restrict__ F2T,
    const float* __restrict__ L1T, const float* __restrict__ L2T,
    const int* __restrict__ edge_index,
    const float* __restrict__ p0, const float* __restrict__ p1,
    const float* __restrict__ x0n, const float* __restrict__ x1n,
    float* __restrict__ acc) {
  __shared__ __align__(16) float s0S[16 * 160];
  __shared__ __align__(16) float hN[16 * 32];
  __shared__ __align__(16) float hE[16 * 32];
  __shared__ int dstS[16];
  __shared__ int srcS[16];

  const int lane = threadIdx.x;
  const int e0 = blockIdx.x * 16;

  __builtin_prefetch(F2T, 0, 1);
  __builtin_prefetch(L2T, 0, 1);

  if (lane < 16) {
    dstS[lane] = edge_index[e0 + lane];
    srcS[lane] = edge_index[N_EDGES + e0 + lane];
  }
  __syncthreads();

  // s0 = [p0[dst] | p0[src] | <p1[dst],p1[src]>/sqrt(3)]  (16 x 160 in LDS)
  for (int idx = lane; idx < 16 * 160; idx += 32) {
    int m = idx / 160, c = idx % 160;
    int d = dstS[m], s = srcS[m];
    float v;
    if (c < 64) {
      v = p0[(size_t)d * 64 + c];
    } else if (c < 128) {
      v = p0[(size_t)s * 64 + (c - 64)];
    } else {
      int u = c - 128;
      const float* pd = p1 + (size_t)d * 96 + u * 3;
      const float* ps = p1 + (size_t)s * 96 + u * 3;
      v = (pd[0] * ps[0] + pd[1] * ps[1] + pd[2] * ps[2]) * INV_SQ3;
    }
    s0S[idx] = v;
  }
  __syncthreads();

  // hN = ssp(s0 @ L1 / sqrt(160))   (K=160)
  {
    v2f as0[40];
    load_a(as0, [&](int m, int k) { return ld2(s0S + m * 160 + k); });
    for (int j = 0; j < 2; ++j) {
      v8f c = {};
      c = wmma_tile(as0, [&](int n, int k) {
        return ld2(L1T + (size_t)(j * 16 + n) * 160 + k);
      }, c);
      tile_store(c, [&](int, int m, int n, float v) {
        hN[m * 32 + j * 16 + n] = sspf(v * INV_SQ160);
      });
    }
  }
  // hE = ssp(edge_attr @ F1 / sqrt(32))   (K=32, A straight from global)
  {
    const float* ea = edge_attr + (size_t)e0 * 32;
    v2f aea[8];
    load_a(aea, [&](int m, int k) { return ld2(ea + m * 32 + k); });
    for (int j = 0; j < 2; ++j) {
      v8f c = {};
      c = wmma_tile(aea, [&](int n, int k) {
        return ld2(F1T + (size_t)(j * 16 + n) * 32 + k);
      }, c);
      tile_store(c, [&](int, int m, int n, float v) {
        hE[m * 32 + j * 16 + n] = sspf(v * INV_SQ32);
      });
    }
  }
  __syncthreads();

  // Hoist per-row edge data: each lane only touches rows rbase..rbase+7.
  const int rbase = (lane >> 4) << 3;
  float sh0r[8], s1xr[8], s1yr[8], s1zr[8];
  int dR[8], sR[8];
#pragma unroll
  for (int r = 0; r < 8; ++r) {
    int m = rbase + r;
    int e = e0 + m;
    sh0r[r] = edge_sh[(size_t)e * 4 + 0];
    s1xr[r] = edge_sh[(size_t)e * 4 + 1];
    s1yr[r] = edge_sh[(size_t)e * 4 + 2];
    s1zr[r] = edge_sh[(size_t)e * 4 + 3];
    dR[r] = dstS[m];
    sR[r] = srcS[m];
  }

  // A panels for the two 32->224 GEMMs, loaded once, reused by 14 tiles.
  v2f aN[8], aE[8];
  load_a(aN, [&](int m, int k) { return ld2(hN + m * 32 + k); });
  load_a(aE, [&](int m, int k) { return ld2(hE + m * 32 + k); });

  for (int j = 0; j < 14; ++j) {
    const int nb = j * 16;
    v8f cN = {}, cE = {};
    cN = wmma_tile(aN, [&](int n, int k) {
      return ld2(L2T + (size_t)(nb + n) * 32 + k);
    }, cN);
    cE = wmma_tile(aE, [&](int n, int k) {
      return ld2(F2T + (size_t)(nb + n) * 32 + k);
    }, cE);
    v8f w8 = cN * cE * (INV_SQ32 * INV_SQ32);

    tile_store(w8, [&](int r, int /*m*/, int n, float wv) {
      const int col = nb + n;
      const float sh0 = sh0r[r], s1x = s1xr[r], s1y = s1yr[r], s1z = s1zr[r];
      const int s = sR[r];
      float* ad = acc + (size_t)dR[r] * 480;
      if (col < 64) {                       // w1: o0a -> ch [0,64)
        float xs = x0n[(size_t)s * 64 + col];
        atomicAdd(&ad[col], wv * xs * sh0);
      } else if (col < 128) {               // w2: o1a -> ch [96,288)
        int c2 = col - 64;
        float f = wv * x0n[(size_t)s * 64 + c2] * INV_SQ3;
        atomicAdd(&ad[96 + c2 * 3 + 0], f * s1x);
        atomicAdd(&ad[96 + c2 * 3 + 1], f * s1y);
        atomicAdd(&ad[96 + c2 * 3 + 2], f * s1z);
      } else if (col < 160) {               // w3: o1b -> ch [288,384)
        int c3 = col - 128;
        const float* xv = x1n + (size_t)s * 96 + c3 * 3;
        float f = wv * sh0;
        atomicAdd(&ad[288 + c3 * 3 + 0], f * xv[0]);
        atomicAdd(&ad[288 + c3 * 3 + 1], f * xv[1]);
        atomicAdd(&ad[288 + c3 * 3 + 2], f * xv[2]);
      } else if (col < 192) {               // w4: o0b -> ch [64,96)
        int c4 = col - 160;
        const float* xv = x1n + (size_t)s * 96 + c4 * 3;
        float dp = xv[0] * s1x + xv[1] * s1y + xv[2] * s1z;
        atomicAdd(&ad[64 + c4], wv * dp * INV_SQ3);
      } else {                              // w5: o1c (cross) -> ch [384,480)
        int c5 = col - 192;
        const float* xv = x1n + (size_t)s * 96 + c5 * 3;
        float cx = xv[1] * s1z - xv[2] * s1y;
        float cy = xv[2] * s1x - xv[0] * s1z;
        float cz = xv[0] * s1y - xv[1] * s1x;
        float f = wv * INV_SQ2;
        atomicAdd(&ad[384 + c5 * 3 + 0], f * cx);
        atomicAdd(&ad[384 + c5 * 3 + 1], f * cy);
        atomicAdd(&ad[384 + c5 * 3 + 2], f * cz);
      }
    });
  }
}

// ---------------------------------------------------------------------------
// Kernel 3: output projections on the accumulated node features.
// ---------------------------------------------------------------------------
__global__ __launch_bounds__(32) void proj_kernel(
    const float* __restrict__ acc,
    const float* __restrict__ WO0T, const float* __restrict__ bO0,
    const float* __restrict__ WO1T, const float* __restrict__ WO2T,
    float* __restrict__ y) {
  const int n0 = blockIdx.x * 16;
  const float* at = acc + (size_t)n0 * 480;

  // y0 = t0 @ WO0/sqrt(96) + bO0  -> y[:, 0:128]   (K=96, 8 tiles)
  {
    v2f aa[24];
    load_a(aa, [&](int m, int k) { return ld2(at + (size_t)m * 480 + k); });
    for (int j = 0; j < 8; ++j) {
      v8f c = {};
      c = wmma_tile(aa, [&](int n, int k) {
        return ld2(WO0T + (size_t)(j * 16 + n) * 96 + k);
      }, c);
      tile_store(c, [&](int, int m, int n, float v) {
        int col = j * 16 + n;
        y[(size_t)(n0 + m) * 416 + col] = v * INV_SQ96 + bO0[col];
      });
    }
  }
  // y1 = einsum(t1, WO1)/sqrt(96) -> y[:, 128:320]   (K=96, strided A)
  for (int i = 0; i < 3; ++i) {
    v2f aa[24];
    load_a(aa, [&](int m, int k) {
      v2f a;
      a.x = at[(size_t)m * 480 + 96 + 3 * k + i];
      a.y = at[(size_t)m * 480 + 96 + 3 * (k + 1) + i];
      return a;
    });
    for (int j = 0; j < 4; ++j) {
      v8f c = {};
      c = wmma_tile(aa, [&](int n, int k) {
        return ld2(WO1T + (size_t)(j * 16 + n) * 96 + k);
      }, c);
      tile_store(c, [&](int, int m, int n, float v) {
        y[(size_t)(n0 + m) * 416 + 128 + (j * 16 + n) * 3 + i] = v * INV_SQ96;
      });
    }
  }
  // y2 = einsum(t2, WO2)/sqrt(32) -> y[:, 320:416]   (K=32, strided A)
  for (int i = 0; i < 3; ++i) {
    v2f aa[8];
    load_a(aa, [&](int m, int k) {
      v2f a;
      a.x = at[(size_t)m * 480 + 384 + 3 * k + i];
      a.y = at[(size_t)m * 480 + 384 + 3 * (k + 1) + i];
      return a;
    });
    for (int j = 0; j < 2; ++j) {
      v8f c = {};
      c = wmma_tile(aa, [&](int n, int k) {
        return ld2(WO2T + (size_t)(j * 16 + n) * 32 + k);
      }, c);
      tile_store(c, [&](int, int m, int n, float v) {
        y[(size_t)(n0 + m) * 416 + 320 + (j * 16 + n) * 3 + i] = v * INV_SQ32;
      });
    }
  }
}

// ---------------------------------------------------------------------------
extern "C" void kernel_launch(void* const* d_in, const int* in_sizes, int n_in,
                              void* d_out, int out_size, void* d_ws,
                              size_t ws_size, hipStream_t stream) {
  (void)in_sizes; (void)n_in; (void)out_size; (void)ws_size;
  const float* x         = (const float*)d_in[0];
  const float* edge_sh   = (const float*)d_in[1];
  const float* edge_attr = (const float*)d_in[2];
  const float* W0p = (const float*)d_in[3];
  const float* b0p = (const float*)d_in[4];
  const float* W1p = (const float*)d_in[5];
  const float* W0n = (const float*)d_in[6];
  const float* b0n = (const float*)d_in[7];
  const float* W1n = (const float*)d_in[8];
  const float* G1  = (const float*)d_in[9];
  const float* g1b = (const float*)d_in[10];
  const float* G2  = (const float*)d_in[11];
  const float* g2b = (const float*)d_in[12];
  const float* F1  = (const float*)d_in[13];
  const float* F2  = (const float*)d_in[14];
  const float* L1  = (const float*)d_in[15];
  const float* L2  = (const float*)d_in[16];
  const float* WO0 = (const float*)d_in[17];
  const float* bO0 = (const float*)d_in[18];
  const float* WO1 = (const float*)d_in[19];
  const float* WO2 = (const float*)d_in[20];
  const int* edge_index = (const int*)d_in[21];

  // Workspace layout (floats): node intermediates + L2-resident accumulator
  // + fragment-major (transposed) weight copies.  Total ~32.3 MB.
  float* ws   = (float*)d_ws;
  float* p0   = ws;                               // N*64
  float* p1   = p0   + (size_t)N_NODES * 64;      // N*96
  float* x0n  = p1   + (size_t)N_NODES * 96;      // N*64
  float* x1n  = x0n  + (size_t)N_NODES * 64;      // N*96
  float* acc  = x1n  + (size_t)N_NODES * 96;      // N*480
  float* wt   = acc  + (size_t)N_NODES * 480;
  float* W0pT = wt;            wt += 64 * 64;
  float* W1pT = wt;            wt += 32 * 32;
  float* W0nT = wt;            wt += 64 * 64;
  float* W1nT = wt;            wt += 32 * 32;
  float* G1T  = wt;            wt += 96 * 96;
  float* G2T  = wt;            wt += 96 * 96;
  float* F1T  = wt;            wt += 32 * 32;
  float* F2T  = wt;            wt += 32 * 224;
  float* L1T  = wt;            wt += 160 * 32;
  float* L2T  = wt;            wt += 32 * 224;
  float* WO0T = wt;            wt += 96 * 128;
  float* WO1T = wt;            wt += 96 * 64;
  float* WO2T = wt;            wt += 32 * 32;

  auto tr = [&](const float* W, float* Wt, int K, int Nc) {
    int n = K * Nc;
    transpose_kernel<<<(n + 255) / 256, 256, 0, stream>>>(W, Wt, K, Nc);
  };
  tr(W0p, W0pT, 64, 64);
  tr(W1p, W1pT, 32, 32);
  tr(W0n, W0nT, 64, 64);
  tr(W1n, W1nT, 32, 32);
  tr(G1, G1T, 96, 96);
  tr(G2, G2T, 96, 96);
  tr(F1, F1T, 32, 32);
  tr(F2, F2T, 32, 224);
  tr(L1, L1T, 160, 32);
  tr(L2, L2T, 32, 224);
  tr(WO0, WO0T, 96, 128);
  tr(WO1, WO1T, 96, 64);
  tr(WO2, WO2T, 32, 32);

  hipMemsetAsync(acc, 0, (size_t)N_NODES * 480 * sizeof(float), stream);

  node_kernel<<<N_NODES / 16, 32, 0, stream>>>(
      x, W0pT, b0p, W1pT, W0nT, b0n, W1nT, G1T, g1b, G2T, g2b,
      p0, p1, x0n, x1n);

  edge_kernel<<<N_EDGES / 16, 32, 0, stream>>>(
      edge_sh, edge_attr, F1T, F2T, L1T, L2T, edge_index,
      p0, p1, x0n, x1n, acc);

  proj_kernel<<<N_NODES / 16, 32, 0, stream>>>(
      acc, WO0T, bO0, WO1T, WO2T, (float*)d_out);
}